// Faceformer_75402445848930
// MI455X (gfx1250) — compile-verified
//
#include <hip/hip_runtime.h>
#include <math.h>

// ---------------------------------------------------------------------------
// Faceformer autoregressive decode for MI455X (gfx1250, wave32).
//
// Exact-equivalent incremental formulation of the reference:
//  - precompute mem = hidden_audio@W_audio+b (WMMA f32 GEMM, M=192,K=768,N=512)
//  - precompute cross K/V = mem@ca_Wk/Wv + b   (WMMA f32 GEMM)
//  - per step i (strictly sequential, stream-ordered launches):
//      k1: decoder row i (self-attn w/ cached K/V + ALiBi, cross-attn w/ diag
//          mask, FF, 3x post-LN)  -> h[B,512]       (512 thr = 16 wave32)
//      k2: vrow = h @ W_vmapr + b  (GEMV, N=15069), writes output row
//      k3: partial K-sums of vrow @ W_vmap (32 chunks, deterministic order);
//          next step's k1 reduces partials + b_vmap + style = buf[i+1]
// Both 31MB vertex-map weight matrices are L2-resident (192MB L2), so the
// per-step GEMV streams hit L2, not HBM.  The GEMV kernels stage their small
// activation vectors into LDS with the Tensor Data Mover (tensor_load_to_lds
// + s_wait_tensorcnt), the CDNA5 async copy path.
// ---------------------------------------------------------------------------

#define D512   512
#define TT     96
#define HH     4
#define EE     128
#define VD     15069
#define NCHUNK 32
#define CHUNKV 471   // 32*471 = 15072 >= 15069
#define NEGBIG (-1e9f)

typedef float v2f __attribute__((ext_vector_type(2)));
typedef float v8f __attribute__((ext_vector_type(8)));
typedef unsigned v4u __attribute__((ext_vector_type(4)));
typedef unsigned v8u __attribute__((ext_vector_type(8)));

// ---------------------------------------------------------------------------
// TDM: 1-D tile load of `nelem` f32 from global -> LDS via the Tensor Data
// Mover (CDNA5 ISA).  Caller: issue from ONE wave of the
// block (TDM issue is per-wave, EXEC ignored), then s_wait_tensorcnt 0 and
// barrier before other waves read the LDS destination.
// D# group0: [1:0]=count=1, [63:32]=lds_addr, [120:64]=global_addr,
//            [127:126]=type=2.
// D# group1: [17:16]=data_size=2 (4B), tensor_dim0=nelem (d1[31:16]|d2[15:0]),
//            tensor_dim1=1, tile_dim0=nelem (d3[31:16]), tile_dim1/2=0 unused,
//            tensor_dim0_stride=nelem.
// ---------------------------------------------------------------------------
__device__ __forceinline__ void tdm_load_1d_f32(void* lds_dst, const void* gsrc,
                                                unsigned nelem)
{
    const unsigned long long ga = (unsigned long long)(uintptr_t)gsrc;
    const unsigned lds = (unsigned)(uintptr_t)lds_dst;  // generic LDS addr: low 32 = offset
    v4u g0;
    g0.x = 1u;                                          // count=1 (valid user D#)
    g0.y = lds;                                         // lds_addr (bytes)
    g0.z = (unsigned)(ga & 0xFFFFFFFFu);                // global_addr[31:0]
    g0.w = (unsigned)((ga >> 32) & 0x01FFFFFFu) | (2u << 30);  // addr[56:32] | type=2
    v8u g1;
    g1.s0 = (2u << 16);                                 // wg_mask=0, data_size=4B
    g1.s1 = (nelem & 0xFFFFu) << 16;                    // tensor_dim0[15:0]
    g1.s2 = ((nelem >> 16) & 0xFFFFu) | (1u << 16);     // tensor_dim0[31:16], tensor_dim1=1
    g1.s3 = (nelem & 0xFFFFu) << 16;                    // tensor_dim1 hi=0, tile_dim0=nelem
    g1.s4 = 0u;                                         // tile_dim1=0, tile_dim2=0 (unused)
    g1.s5 = nelem;                                      // tensor_dim0_stride[31:0]
    g1.s6 = 0u;                                         // stride hi / dim1_stride lo
    g1.s7 = 0u;
    asm volatile("tensor_load_to_lds %0, %1" :: "s"(g0), "s"(g1) : "memory");
}

// ---------------------------------------------------------------------------
// f32 WMMA GEMM: C[M,N] = A[M,K] @ B[K,N] + bias.  One 16x16 tile per wave,
// 4 waves (128 thr) per block covering 16(M) x 64(N).  M%16==0, N%16==0, K%4==0.
// Uses V_WMMA_F32_16X16X4_F32 (full f32 precision, matches reference).
// ---------------------------------------------------------------------------
__global__ __launch_bounds__(128) void gemm_wmma_f32(
    const float* __restrict__ A, int lda,
    const float* __restrict__ B, int ldb,
    const float* __restrict__ bias,
    float* __restrict__ C, int ldc, int K)
{
    const int lane = threadIdx.x & 31;
    const int wave = threadIdx.x >> 5;
    const int tm   = blockIdx.y * 16;
    const int tn   = (blockIdx.x * 4 + wave) * 16;
    const int rc   = lane & 15;           // row (A) / col (B)
    const int klo  = (lane >> 4) << 1;    // 0 or 2

    v8f acc = {};
    const float* Ap = A + (tm + rc) * lda + klo;
    const float* Bp = B + klo * ldb + (tn + rc);
    for (int k0 = 0; k0 < K; k0 += 4) {
        v2f av, bv;
        av.x = Ap[0];
        av.y = Ap[1];
        bv.x = Bp[0];
        bv.y = Bp[ldb];
        acc = __builtin_amdgcn_wmma_f32_16x16x4_f32(
            /*neg_a=*/false, av, /*neg_b=*/false, bv,
            /*c_mod=*/(short)0, acc, /*reuse_a=*/false, /*reuse_b=*/false);
        Ap += 4;
        Bp += 4 * ldb;
    }
    const int cm = tm + ((lane >> 4) << 3);
    const int cn = tn + (lane & 15);
    const float bs = bias[cn];
    #pragma unroll
    for (int r = 0; r < 8; ++r)
        C[(cm + r) * ldc + cn] = acc[r] + bs;
}

// ---------------------------------------------------------------------------
// small precompute kernels
// ---------------------------------------------------------------------------
__global__ __launch_bounds__(256) void pe_kernel(float* __restrict__ pe)
{
    const int t = blockIdx.x;
    const int pos = t % 30;
    for (int d = threadIdx.x; d < D512; d += 256) {
        const int k = d >> 1;
        const float inv = expf((float)(2 * k) * (-9.210340371976184f / 512.0f));
        const float ang = (float)pos * inv;
        pe[t * D512 + d] = (d & 1) ? cosf(ang) : sinf(ang);
    }
}

__global__ __launch_bounds__(256) void style_kernel(
    const float* __restrict__ one_hot, const float* __restrict__ W_obj,
    float* __restrict__ style)
{
    const int b = blockIdx.x;
    for (int d = threadIdx.x; d < D512; d += 256) {
        float s = 0.f;
        #pragma unroll
        for (int k = 0; k < 8; ++k) s += one_hot[b * 8 + k] * W_obj[k * D512 + d];
        style[b * D512 + d] = s;
    }
}

// ---------------------------------------------------------------------------
// block helpers (blockDim.x == 512)
// ---------------------------------------------------------------------------
__device__ __forceinline__ float block_sum_512t(float v, float* red)
{
    const int tid = threadIdx.x;
    red[tid] = v;
    __syncthreads();
    for (int s = 256; s > 0; s >>= 1) {
        if (tid < s) red[tid] += red[tid + s];
        __syncthreads();
    }
    const float r = red[0];
    __syncthreads();
    return r;
}

__device__ __forceinline__ void layer_norm_512(
    const float* __restrict__ in, float* __restrict__ out,
    const float* __restrict__ g, const float* __restrict__ bb, float* red)
{
    const int tid = threadIdx.x;
    float s = 0.f, sq = 0.f;
    for (int d = tid; d < D512; d += 512) { const float v = in[d]; s += v; sq += v * v; }
    s  = block_sum_512t(s,  red);
    sq = block_sum_512t(sq, red);
    const float mu  = s * (1.0f / 512.0f);
    const float inv = rsqrtf(sq * (1.0f / 512.0f) - mu * mu + 1e-5f);
    for (int d = tid; d < D512; d += 512)
        out[d] = (in[d] - mu) * inv * g[d] + bb[d];
    __syncthreads();
}

// ---------------------------------------------------------------------------
// k1: one decoder step for row i.  grid = B, block = 512 (16 wave32).
// ---------------------------------------------------------------------------
__global__ __launch_bounds__(512) void step_decoder(
    int i,
    const float* __restrict__ partials, // [B,NCHUNK,512] from prev step (i>0)
    const float* __restrict__ style,    // [B,512]
    const float* __restrict__ pe,       // [96,512]
    const float* __restrict__ b_vmap,   // [512]
    const float* __restrict__ sa_Wq, const float* __restrict__ sa_bq,
    const float* __restrict__ sa_Wk, const float* __restrict__ sa_bk,
    const float* __restrict__ sa_Wv, const float* __restrict__ sa_bv,
    const float* __restrict__ sa_Wo, const float* __restrict__ sa_bo,
    const float* __restrict__ ca_Wq, const float* __restrict__ ca_bq,
    const float* __restrict__ ca_Wo, const float* __restrict__ ca_bo,
    const float* __restrict__ W_ff1, const float* __restrict__ b_ff1,
    const float* __restrict__ W_ff2, const float* __restrict__ b_ff2,
    const float* __restrict__ ln1_g, const float* __restrict__ ln1_b,
    const float* __restrict__ ln2_g, const float* __restrict__ ln2_b,
    const float* __restrict__ ln3_g, const float* __restrict__ ln3_b,
    const float* __restrict__ cK,    const float* __restrict__ cV,   // [B*96,512]
    float* __restrict__ selfK, float* __restrict__ selfV,            // [B*96,512]
    float* __restrict__ h_out)                                       // [B,512]
{
    const int b   = blockIdx.x;
    const int tid = threadIdx.x;
    const float scale = 0.08838834764831845f;  // 1/sqrt(128)

    __shared__ float x[D512], qv[D512], attn[D512], t0[D512], t1[1024];
    __shared__ float sc[HH * TT];
    __shared__ float hinv[HH];
    __shared__ float red[512];

    // ---- decoder input row i: buf[i] + pe[i] ----
    for (int d = tid; d < D512; d += 512) {
        float v = style[b * D512 + d];
        if (i > 0) {
            v += b_vmap[d];
            #pragma unroll 4
            for (int c = 0; c < NCHUNK; ++c)
                v += partials[(b * NCHUNK + c) * D512 + d];
        }
        x[d] = v + pe[i * D512 + d];
    }
    __syncthreads();

    // ---- self-attn QKV (M=1 GEMVs), append K/V to cache ----
    for (int d = tid; d < D512; d += 512) {
        float q = sa_bq[d], kk2 = sa_bk[d], vv = sa_bv[d];
        for (int kk = 0; kk < D512; ++kk) {
            const float xv = x[kk];
            q   += xv * sa_Wq[kk * D512 + d];
            kk2 += xv * sa_Wk[kk * D512 + d];
            vv  += xv * sa_Wv[kk * D512 + d];
        }
        qv[d] = q;
        selfK[(b * TT + i) * D512 + d] = kk2;
        selfV[(b * TT + i) * D512 + d] = vv;
    }
    __threadfence_block();
    __syncthreads();

    // ---- self-attn scores with ALiBi (j<=i) ----
    for (int idx = tid; idx < HH * (i + 1); idx += 512) {
        const int h = idx / (i + 1);
        const int j = idx % (i + 1);
        const float* kr = &selfK[(b * TT + j) * D512 + h * EE];
        const float* qr = &qv[h * EE];
        float s = 0.f;
        for (int e = 0; e < EE; ++e) s += qr[e] * kr[e];
        const float slope = exp2f(-2.0f * (float)(h + 1));  // 0.25^(h+1)
        s += slope * (float)(-((i - j) / 30));
        sc[h * TT + j] = s * scale;
    }
    __syncthreads();
    if (tid < HH) {
        float m = -3.4e38f;
        for (int j = 0; j <= i; ++j) m = fmaxf(m, sc[tid * TT + j]);
        float s = 0.f;
        for (int j = 0; j <= i; ++j) {
            const float e = expf(sc[tid * TT + j] - m);
            sc[tid * TT + j] = e;
            s += e;
        }
        hinv[tid] = 1.f / s;
    }
    __syncthreads();
    for (int d = tid; d < D512; d += 512) {
        const int h = d >> 7;
        float o = 0.f;
        for (int j = 0; j <= i; ++j)
            o += sc[h * TT + j] * selfV[(b * TT + j) * D512 + d];
        attn[d] = o * hinv[h];
    }
    __syncthreads();

    // ---- out proj + residual + LN1 ----
    for (int d = tid; d < D512; d += 512) {
        float o = sa_bo[d];
        for (int kk = 0; kk < D512; ++kk) o += attn[kk] * sa_Wo[kk * D512 + d];
        t0[d] = x[d] + o;
    }
    __syncthreads();
    layer_norm_512(t0, x, ln1_g, ln1_b, red);

    // ---- cross-attn: Q from x, K/V precomputed from mem ----
    for (int d = tid; d < D512; d += 512) {
        float q = ca_bq[d];
        for (int kk = 0; kk < D512; ++kk) q += x[kk] * ca_Wq[kk * D512 + d];
        qv[d] = q;
    }
    __syncthreads();
    for (int idx = tid; idx < HH * TT; idx += 512) {
        const int h = idx / TT;
        const int j = idx % TT;
        const float* kr = &cK[(b * TT + j) * D512 + h * EE];
        float s = 0.f;
        for (int e = 0; e < EE; ++e) s += qv[h * EE + e] * kr[e];
        if (j == i) s += NEGBIG;   // diagonal mask, added before temperature
        sc[h * TT + j] = s * scale;
    }
    __syncthreads();
    if (tid < HH) {
        float m = -3.4e38f;
        for (int j = 0; j < TT; ++j) m = fmaxf(m, sc[tid * TT + j]);
        float s = 0.f;
        for (int j = 0; j < TT; ++j) {
            const float e = expf(sc[tid * TT + j] - m);
            sc[tid * TT + j] = e;
            s += e;
        }
        hinv[tid] = 1.f / s;
    }
    __syncthreads();
    for (int d = tid; d < D512; d += 512) {
        const int h = d >> 7;
        float o = 0.f;
        for (int j = 0; j < TT; ++j)
            o += sc[h * TT + j] * cV[(b * TT + j) * D512 + d];
        attn[d] = o * hinv[h];
    }
    __syncthreads();
    for (int d = tid; d < D512; d += 512) {
        float o = ca_bo[d];
        for (int kk = 0; kk < D512; ++kk) o += attn[kk] * ca_Wo[kk * D512 + d];
        t0[d] = x[d] + o;
    }
    __syncthreads();
    layer_norm_512(t0, x, ln2_g, ln2_b, red);

    // ---- FFN: relu(x@W1+b1)@W2+b2, residual, LN3 ----
    for (int dd = tid; dd < 1024; dd += 512) {
        float y = b_ff1[dd];
        for (int kk = 0; kk < D512; ++kk) y += x[kk] * W_ff1[kk * 1024 + dd];
        t1[dd] = fmaxf(y, 0.f);
    }
    __syncthreads();
    for (int d = tid; d < D512; d += 512) {
        float y = b_ff2[d];
        for (int kk = 0; kk < 1024; ++kk) y += t1[kk] * W_ff2[kk * D512 + d];
        t0[d] = x[d] + y;
    }
    __syncthreads();
    layer_norm_512(t0, x, ln3_g, ln3_b, red);

    for (int d = tid; d < D512; d += 512)
        h_out[b * D512 + d] = x[d];
}

// ---------------------------------------------------------------------------
// k2: vrow = h @ W_vmapr + b_vmapr ; out row = vrow + template
// grid = (ceil(VD/256), B), block 256.  h row staged into LDS by the TDM.
// ---------------------------------------------------------------------------
__global__ __launch_bounds__(256) void vmapr_kernel(
    int i,
    const float* __restrict__ h, const float* __restrict__ W_vmapr,
    const float* __restrict__ b_vmapr, const float* __restrict__ tmpl,
    float* __restrict__ vrow, float* __restrict__ out)
{
    const int b = blockIdx.y;
    const int v = blockIdx.x * 256 + threadIdx.x;
    __shared__ float hs[D512];
    if ((threadIdx.x >> 5) == 0) {          // wave 0 issues the TDM copy
        tdm_load_1d_f32(hs, h + b * D512, D512);
        __builtin_amdgcn_s_wait_tensorcnt(0);
    }
    __syncthreads();
    if (v < VD) {
        float s = b_vmapr[v];
        for (int k = 0; k < D512; ++k) {
            if ((k & 63) == 0)
                __builtin_prefetch(&W_vmapr[(k + 64) * VD + v], 0, 0);
            s += hs[k] * W_vmapr[k * VD + v];
        }
        vrow[b * VD + v] = s;
        out[(b * TT + i) * VD + v] = s + tmpl[b * VD + v];
    }
}

// ---------------------------------------------------------------------------
// k3: deterministic split-K partial sums of vrow @ W_vmap.
// grid = (NCHUNK, B), block 256; thread covers d and d+256.
// vrow chunk staged into LDS by the TDM.
// ---------------------------------------------------------------------------
__global__ __launch_bounds__(256) void vmap_partial(
    const float* __restrict__ vrow, const float* __restrict__ W_vmap,
    float* __restrict__ partials)
{
    const int b = blockIdx.y, c = blockIdx.x, tid = threadIdx.x;
    const int v0 = c * CHUNKV;
    int v1 = v0 + CHUNKV; if (v1 > VD) v1 = VD;
    const int nv = v1 - v0;
    __shared__ float vr[CHUNKV];
    if ((tid >> 5) == 0) {                  // wave 0 issues the TDM copy
        tdm_load_1d_f32(vr, vrow + b * VD + v0, (unsigned)nv);
        __builtin_amdgcn_s_wait_tensorcnt(0);
    }
    __syncthreads();
    const int d0 = tid, d1 = tid + 256;
    float a0 = 0.f, a1 = 0.f;
    for (int v = v0; v < v1; ++v) {
        if ((v & 31) == 0)
            __builtin_prefetch(&W_vmap[(v + 32) * D512 + d0], 0, 0);
        const float r = vr[v - v0];
        const float* wr = &W_vmap[v * D512];
        a0 += r * wr[d0];
        a1 += r * wr[d1];
    }
    partials[(b * NCHUNK + c) * D512 + d0] = a0;
    partials[(b * NCHUNK + c) * D512 + d1] = a1;
}

// ---------------------------------------------------------------------------
extern "C" void kernel_launch(void* const* d_in, const int* in_sizes, int n_in,
                              void* d_out, int out_size, void* d_ws, size_t ws_size,
                              hipStream_t stream)
{
    const float* hidden_audio = (const float*)d_in[0];   // [2,96,768]
    const float* one_hot      = (const float*)d_in[1];   // [2,8]
    const float* tmpl         = (const float*)d_in[2];   // [2,15069]
    const float* W_audio      = (const float*)d_in[3];   // [768,512]
    const float* b_audio      = (const float*)d_in[4];
    const float* W_obj        = (const float*)d_in[5];   // [8,512]
    const float* W_vmap       = (const float*)d_in[6];   // [15069,512]
    const float* b_vmap       = (const float*)d_in[7];
    const float* W_vmapr      = (const float*)d_in[8];   // [512,15069]
    const float* b_vmapr      = (const float*)d_in[9];
    const float* sa_Wq = (const float*)d_in[10]; const float* sa_bq = (const float*)d_in[11];
    const float* sa_Wk = (const float*)d_in[12]; const float* sa_bk = (const float*)d_in[13];
    const float* sa_Wv = (const float*)d_in[14]; const float* sa_bv = (const float*)d_in[15];
    const float* sa_Wo = (const float*)d_in[16]; const float* sa_bo = (const float*)d_in[17];
    const float* ca_Wq = (const float*)d_in[18]; const float* ca_bq = (const float*)d_in[19];
    const float* ca_Wk = (const float*)d_in[20]; const float* ca_bk = (const float*)d_in[21];
    const float* ca_Wv = (const float*)d_in[22]; const float* ca_bv = (const float*)d_in[23];
    const float* ca_Wo = (const float*)d_in[24]; const float* ca_bo = (const float*)d_in[25];
    const float* W_ff1 = (const float*)d_in[26]; const float* b_ff1 = (const float*)d_in[27];
    const float* W_ff2 = (const float*)d_in[28]; const float* b_ff2 = (const float*)d_in[29];
    const float* ln1_g = (const float*)d_in[30]; const float* ln1_b = (const float*)d_in[31];
    const float* ln2_g = (const float*)d_in[32]; const float* ln2_b = (const float*)d_in[33];
    const float* ln3_g = (const float*)d_in[34]; const float* ln3_b = (const float*)d_in[35];
    float* out = (float*)d_out;

    // workspace layout (floats)
    float* ws       = (float*)d_ws;
    float* mem      = ws;                     // 192*512
    float* cK       = mem   + 192 * 512;      // 192*512
    float* cV       = cK    + 192 * 512;      // 192*512
    float* pe       = cV    + 192 * 512;      // 96*512
    float* style    = pe    + 96 * 512;       // 2*512
    float* selfK    = style + 2 * 512;        // 192*512
    float* selfV    = selfK + 192 * 512;      // 192*512
    float* hbuf     = selfV + 192 * 512;      // 2*512
    float* vrow     = hbuf  + 2 * 512;        // 2*VD
    float* partials = vrow  + 2 * VD;         // 2*NCHUNK*512

    // ---- precompute ----
    pe_kernel<<<TT, 256, 0, stream>>>(pe);
    style_kernel<<<2, 256, 0, stream>>>(one_hot, W_obj, style);
    // mem = hidden_audio @ W_audio + b_audio   [192,768]x[768,512]
    gemm_wmma_f32<<<dim3(8, 12), 128, 0, stream>>>(hidden_audio, 768, W_audio, 512,
                                                   b_audio, mem, 512, 768);
    // cross K/V = mem @ ca_Wk/Wv + bias        [192,512]x[512,512]
    gemm_wmma_f32<<<dim3(8, 12), 128, 0, stream>>>(mem, 512, ca_Wk, 512,
                                                   ca_bk, cK, 512, 512);
    gemm_wmma_f32<<<dim3(8, 12), 128, 0, stream>>>(mem, 512, ca_Wv, 512,
                                                   ca_bv, cV, 512, 512);

    // ---- autoregressive loop (sequential dependency) ----
    for (int i = 0; i < TT; ++i) {
        step_decoder<<<2, 512, 0, stream>>>(
            i, partials, style, pe, b_vmap,
            sa_Wq, sa_bq, sa_Wk, sa_bk, sa_Wv, sa_bv, sa_Wo, sa_bo,
            ca_Wq, ca_bq, ca_Wo, ca_bo,
            W_ff1, b_ff1, W_ff2, b_ff2,
            ln1_g, ln1_b, ln2_g, ln2_b, ln3_g, ln3_b,
            cK, cV, selfK, selfV, hbuf);
        vmapr_kernel<<<dim3((VD + 255) / 256, 2), 256, 0, stream>>>(
            i, hbuf, W_vmapr, b_vmapr, tmpl, vrow, out);
        if (i + 1 < TT)
            vmap_partial<<<dim3(NCHUNK, 2), 256, 0, stream>>>(vrow, W_vmap, partials);
    }
}